// Generator_90142773608853
// MI455X (gfx1250) — compile-verified
//
#include <hip/hip_runtime.h>
#include <hip/hip_bf16.h>
#include <math.h>

// ---------------------------------------------------------------------------
// Problem constants (from reference)
// ---------------------------------------------------------------------------
#define NB    4        // batch
#define LF    256      // frame length
#define CF    256      // ConvNeXt width
#define HF    512      // ConvNeXt hidden
#define NHARM 48
#define SEG   960
#define LW    (LF * SEG)   // 245760
#define SRF   48000.0f
#define NLAY  6

typedef float v2f __attribute__((ext_vector_type(2)));
typedef float v8f __attribute__((ext_vector_type(8)));

__device__ __forceinline__ v8f wmma4(v2f a, v2f b, v8f c) {
    // D = A(16x4,f32) * B(4x16,f32) + C(16x16,f32)  -- V_WMMA_F32_16X16X4_F32
    return __builtin_amdgcn_wmma_f32_16x16x4_f32(false, a, false, b, (short)0, c,
                                                 false, false);
}

// ---------------------------------------------------------------------------
// Generic WMMA fp32 GEMM over feature frames:
//   Y[n][m][l] = act( sum_k W[m][k] * X[n][k][l] + bias[m] )   (optional res)
// One wave (32 threads) per 16x16 (m,l) tile. L == 256 columns, batch z.
// ACT: 0 = none, 1 = exact GELU, 2 = exp(min(x,4)).  K is compile-time.
// ---------------------------------------------------------------------------
template <int ACT, bool RES, int K>
__global__ void gemm_wmma_k(const float* __restrict__ W,
                            const float* __restrict__ bias,
                            const float* __restrict__ X,
                            float* __restrict__ Y,
                            const float* __restrict__ res,
                            const float* __restrict__ scale,
                            int M) {
    const int lane = threadIdx.x;           // 32 threads / block (one wave)
    const int tm   = blockIdx.x;            // M/16 tiles
    const int tl   = blockIdx.y;            // 16 tiles of L=256
    const int n    = blockIdx.z;            // batch
    const int m16  = lane & 15;
    const int lcol = tl * 16 + m16;
    const int koff = (lane < 16) ? 0 : 2;

    const float* Wr = W + (tm * 16 + m16) * K + koff;           // a.x at +k0
    const float* Xc = X + ((size_t)(n * K) + koff) * LF + lcol; // b.x at +k0*LF

    v8f c = {};
#pragma unroll
    for (int k0 = 0; k0 < K; k0 += 4) {
        v2f a, b;
        a.x = Wr[k0];
        a.y = Wr[k0 + 1];
        b.x = Xc[(size_t)k0 * LF];
        b.y = Xc[(size_t)(k0 + 1) * LF];
        c = wmma4(a, b, c);
    }

    const int mbase = tm * 16 + ((lane < 16) ? 0 : 8);
#pragma unroll
    for (int r = 0; r < 8; ++r) {
        const int m = mbase + r;
        float v = c[r] + bias[m];
        if (ACT == 1) v = 0.5f * v * (1.0f + erff(v * 0.70710678118654752f));
        else if (ACT == 2) v = expf(fminf(v, 4.0f));
        if (RES) v = res[((size_t)(n * M) + m) * LF + lcol] + scale[m] * v;
        Y[((size_t)(n * M) + m) * LF + lcol] = v;
    }
}

// ---------------------------------------------------------------------------
// Depthwise causal conv (K=7) + channel norm. Block = 256 threads (channels),
// grid = (L, N).
// ---------------------------------------------------------------------------
__global__ void dwnorm_k(const float* __restrict__ x,
                         float* __restrict__ out,
                         const float* __restrict__ dww,  // [256][7]
                         const float* __restrict__ dwb,  // [256]
                         const float* __restrict__ ns,   // [256]
                         const float* __restrict__ nb) { // [256]
    const int l = blockIdx.x, n = blockIdx.y, cch = threadIdx.x;
    const float* xr = x + ((size_t)(n * CF) + cch) * LF;
    float acc = dwb[cch];
#pragma unroll
    for (int t = 0; t < 7; ++t) {
        int ll = l - 6 + t;
        if (ll >= 0) acc += dww[cch * 7 + t] * xr[ll];
    }
    __shared__ float red[256];
    red[cch] = acc;
    __syncthreads();
    for (int off = 128; off > 0; off >>= 1) {
        if (cch < off) red[cch] += red[cch + off];
        __syncthreads();
    }
    const float mu = red[0] * (1.0f / 256.0f);
    __syncthreads();
    red[cch] = acc * acc;
    __syncthreads();
    for (int off = 128; off > 0; off >>= 1) {
        if (cch < off) red[cch] += red[cch + off];
        __syncthreads();
    }
    const float var = red[0] * (1.0f / 256.0f) - mu * mu;
    const float h = (acc - mu) * rsqrtf(var + 1e-4f) * ns[cch] + nb[cch];
    out[((size_t)(n * CF) + cch) * LF + l] = h;
}

// ---------------------------------------------------------------------------
// f0 upsample helper (linear interp, align_corners=False)
// ---------------------------------------------------------------------------
__device__ __forceinline__ float f0_up(const float* __restrict__ f0n, int j) {
    float pos = ((float)j + 0.5f) * (1.0f / (float)SEG) - 0.5f;
    pos = fminf(fmaxf(pos, 0.0f), 255.0f);
    const float fi = floorf(pos);
    const int i0 = (int)fi;
    const int i1 = min(i0 + 1, 255);
    const float w = pos - fi;
    return f0n[i0] * (1.0f - w) + f0n[i1] * w;
}

// Chunk sums: grid (256, NB), block 256, chunk = 960 samples
__global__ void f0_chunksum_k(const float* __restrict__ f0, float* __restrict__ csum) {
    const int s = blockIdx.x, n = blockIdx.y, t = threadIdx.x;
    const float* f0n = f0 + n * LF;
    float loc = 0.0f;
    if (t < 240) {
        const int jb = s * SEG + t * 4;
#pragma unroll
        for (int q = 0; q < 4; ++q) loc += f0_up(f0n, jb + q);
    }
    __shared__ float red[256];
    red[t] = loc;
    __syncthreads();
    for (int off = 128; off > 0; off >>= 1) {
        if (t < off) red[t] += red[t + off];
        __syncthreads();
    }
    if (t == 0) csum[n * 256 + s] = red[0];
}

// Exclusive prefix over 256 chunk sums per batch. grid NB, block 256.
__global__ void f0_prefix_k(const float* __restrict__ csum, float* __restrict__ cpre) {
    const int n = blockIdx.x, t = threadIdx.x;
    __shared__ float sc[256];
    const float v = csum[n * 256 + t];
    sc[t] = v;
    __syncthreads();
    for (int off = 1; off < 256; off <<= 1) {
        const float cur = sc[t];
        const float add = (t >= off) ? sc[t - off] : 0.0f;
        __syncthreads();
        sc[t] = cur + add;
        __syncthreads();
    }
    cpre[n * 256 + t] = sc[t] - v;   // exclusive
}

// ---------------------------------------------------------------------------
// Harmonic synthesis: grid (256 chunks, NB), block 256.
// In-chunk scan of f0_up, phase = cumsum/SR, 48 harmonics via sin recurrence.
// ---------------------------------------------------------------------------
__global__ void synth_k(const float* __restrict__ f0,
                        const float* __restrict__ mag,   // [NB][48][256]
                        const float* __restrict__ cpre,
                        float* __restrict__ wave) {
    const int s = blockIdx.x, n = blockIdx.y, t = threadIdx.x;
    __shared__ float smag[NHARM * 4];   // frames base..base+2, pitch 4
    __shared__ float scn[256];
    const int base = (s == 0) ? 0 : (s - 1);
    for (int idx = t; idx < NHARM * 3; idx += 256) {
        const int k = idx / 3, r = idx - k * 3;
        smag[k * 4 + r] = mag[((size_t)(n * NHARM) + k) * LF + min(base + r, 255)];
    }
    const float* f0n = f0 + n * LF;
    float s4[4];
    float loc = 0.0f;
    const bool valid = (t < 240);
    const int jb = s * SEG + t * 4;
    if (valid) {
#pragma unroll
        for (int q = 0; q < 4; ++q) { s4[q] = f0_up(f0n, jb + q); loc += s4[q]; }
    }
    scn[t] = loc;
    __syncthreads();
    for (int off = 1; off < 256; off <<= 1) {
        const float cur = scn[t];
        const float add = (t >= off) ? scn[t - off] : 0.0f;
        __syncthreads();
        scn[t] = cur + add;
        __syncthreads();
    }
    if (!valid) return;
    float run = cpre[n * 256 + s] + (scn[t] - loc);
#pragma unroll
    for (int q = 0; q < 4; ++q) {
        const int j = jb + q;
        run += s4[q];
        const float ph = run * (1.0f / SRF);
        const float th = 6.28318530717958648f * (ph - floorf(ph));
        float sn, cn;
        __sincosf(th, &sn, &cn);
        float pos = ((float)j + 0.5f) * (1.0f / (float)SEG) - 0.5f;
        pos = fminf(fmaxf(pos, 0.0f), 255.0f);
        const float fi = floorf(pos);
        const int i0 = (int)fi, i1 = min(i0 + 1, 255);
        const float w = pos - fi;
        const int r0 = i0 - base, r1 = i1 - base;
        const float twoc = 2.0f * cn;
        float sprev = 0.0f, scur = sn, acc = 0.0f;
        for (int k = 0; k < NHARM; ++k) {
            const float mv = smag[k * 4 + r0] * (1.0f - w) + smag[k * 4 + r1] * w;
            acc += scur * mv;
            const float snext = twoc * scur - sprev;
            sprev = scur; scur = snext;
        }
        wave[(size_t)n * LW + j] = acc * (1.0f / (float)NHARM);
    }
}

// ---------------------------------------------------------------------------
// Reverb input conv 1->16, K=10, causal. One thread per (n,c,j).
// (Reference applies leaky_relu as a no-op here -> no activation.)
// ---------------------------------------------------------------------------
__global__ void rv_in_k(const float* __restrict__ wave,
                        const float* __restrict__ w,   // [16][1][10]
                        const float* __restrict__ b,   // [16]
                        float* __restrict__ out) {
    const size_t idx = (size_t)blockIdx.x * 256 + threadIdx.x;
    const int j = (int)(idx % LW);
    const int c = (int)((idx / LW) & 15);
    const int n = (int)(idx / ((size_t)16 * LW));
    const float* wn = wave + (size_t)n * LW;
    float acc = b[c];
#pragma unroll
    for (int t = 0; t < 10; ++t) {
        const int jj = j - 9 + t;
        if (jj >= 0) acc += w[c * 10 + t] * wn[jj];
    }
    out[((size_t)(n * 16) + c) * LW + j] = acc;
}

// ---------------------------------------------------------------------------
// Dilated reverb layer as implicit GEMM with WMMA f32 16x16x4.
//   out[c][j] = lrelu( b[c] + sum_{i,t} W[c][i][t] * in[i][j-(9-t)*d] )
// M=16 channels, K=160 with permuted order k = t*16 + i so every LDS address
// in the fully unrolled loop is an immediate offset (no runtime div/mul).
// Dilation D is a template constant. Block = 8 waves, 128 samples per block.
// ---------------------------------------------------------------------------
#define RTS 128
#define SPITCH 208   // >= 9*8 + 128 = 200
template <int D>
__global__ void reverb_wmma_k(const float* __restrict__ in,
                              float* __restrict__ out,
                              const float* __restrict__ Wg,   // [16][16][10]
                              const float* __restrict__ bias) {
    __shared__ float sWp[16 * 160];      // permuted: [m][t*16+i]
    __shared__ float sB[16];
    __shared__ float sIn[16 * SPITCH];
    const int tid = threadIdx.x;
    const int n = blockIdx.y;
    const int jBase = blockIdx.x * RTS;
    const int hw = 9 * D;
    const int width = hw + RTS;          // compile-time constant

    // stage + permute weights: sWp[m*160 + t*16 + i] = W[m][i][t]
    for (int idx = tid; idx < 16 * 160; idx += 256) {
        const int m = idx / 160;
        const int r = idx - m * 160;
        const int i = r / 10;
        const int t = r - i * 10;
        sWp[m * 160 + t * 16 + i] = Wg[idx];
    }
    if (tid < 16) sB[tid] = bias[tid];
    // stage input tile (with causal zero halo)
    for (int idx = tid; idx < 16 * width; idx += 256) {
        const int ch = idx / width;
        const int x  = idx - ch * width;
        const int g  = jBase - hw + x;
        sIn[ch * SPITCH + x] = (g >= 0) ? in[((size_t)(n * 16) + ch) * LW + g] : 0.0f;
    }
    __syncthreads();

    const int lane = tid & 31;
    const int waveoff = (tid >> 5) * 16;
    const int nn = lane & 15;
    const int koff = (lane < 16) ? 0 : 2;
    const int mbase = (lane < 16) ? 0 : 8;

    v8f c = {};
#pragma unroll
    for (int r = 0; r < 8; ++r) c[r] = sB[r + mbase];   // bias in accumulator

    // With k = t*16 + i and k0 stepping by 4:  t = k0>>4 (no carry from koff),
    // i = (k0&15) + koff (+1 for the .y element). All offsets are immediates.
    const float* ap = sWp + nn * 160 + koff;
    const float* bp = sIn + koff * SPITCH + (waveoff + nn);
#pragma unroll
    for (int k0 = 0; k0 < 160; k0 += 4) {
        const int t  = k0 >> 4;
        const int ib = k0 & 15;
        v2f a, b;
        a.x = ap[k0];
        a.y = ap[k0 + 1];
        b.x = bp[ib * SPITCH + t * D];
        b.y = bp[(ib + 1) * SPITCH + t * D];
        c = wmma4(a, b, c);
    }

    const int j = jBase + waveoff + nn;
#pragma unroll
    for (int r = 0; r < 8; ++r) {
        float v = c[r];
        v = (v >= 0.0f) ? v : 0.1f * v;                 // leaky relu 0.1
        out[((size_t)(n * 16) + (r + mbase)) * LW + j] = v;
    }
}

// Reverb output conv 16->1, K=10. One thread per (n,j); writes d_out.
__global__ void rv_out_k(const float* __restrict__ in,
                         const float* __restrict__ w,   // [1][16][10]
                         const float* __restrict__ b,   // [1]
                         float* __restrict__ out) {
    const size_t idx = (size_t)blockIdx.x * 256 + threadIdx.x;
    const int j = (int)(idx % LW);
    const int n = (int)(idx / LW);
    float acc = b[0];
#pragma unroll
    for (int cch = 0; cch < 16; ++cch) {
        const float* row = in + ((size_t)(n * 16) + cch) * LW;
#pragma unroll
        for (int t = 0; t < 10; ++t) {
            const int jj = j - 9 + t;
            if (jj >= 0) acc += w[cch * 10 + t] * row[jj];
        }
    }
    out[(size_t)n * LW + j] = acc;
}

// Mean subtraction (deterministic two-level reduction, no atomics)
__global__ void mean_part_k(const float* __restrict__ r, float* __restrict__ part) {
    const int blk = blockIdx.x, n = blockIdx.y, t = threadIdx.x;   // 240 blocks x 256
    const size_t base = (size_t)n * LW + (size_t)blk * 1024;
    float acc = 0.0f;
#pragma unroll
    for (int q = 0; q < 4; ++q) acc += r[base + t + q * 256];
    __shared__ float red[256];
    red[t] = acc;
    __syncthreads();
    for (int off = 128; off > 0; off >>= 1) {
        if (t < off) red[t] += red[t + off];
        __syncthreads();
    }
    if (t == 0) part[n * 240 + blk] = red[0];
}

__global__ void mean_final_k(const float* __restrict__ part, float* __restrict__ mean) {
    const int n = blockIdx.x, t = threadIdx.x;
    __shared__ float red[256];
    red[t] = (t < 240) ? part[n * 240 + t] : 0.0f;
    __syncthreads();
    for (int off = 128; off > 0; off >>= 1) {
        if (t < off) red[t] += red[t + off];
        __syncthreads();
    }
    if (t == 0) mean[n] = red[0] * (1.0f / (float)LW);
}

__global__ void sub_mean_k(float* __restrict__ r, const float* __restrict__ mean) {
    const size_t i = (size_t)blockIdx.x * 256 + threadIdx.x;
    const int n = (int)(i / LW);
    r[i] -= mean[n];
}

// ---------------------------------------------------------------------------
// Launch
// ---------------------------------------------------------------------------
extern "C" void kernel_launch(void* const* d_in, const int* in_sizes, int n_in,
                              void* d_out, int out_size, void* d_ws, size_t ws_size,
                              hipStream_t stream) {
    const float* mel      = (const float*)d_in[0];
    const float* f0       = (const float*)d_in[1];
    const float* fe_in_w  = (const float*)d_in[2];
    const float* fe_in_b  = (const float*)d_in[3];
    const float* fe_dw_w  = (const float*)d_in[4];
    const float* fe_dw_b  = (const float*)d_in[5];
    const float* fe_ns    = (const float*)d_in[6];
    const float* fe_nb    = (const float*)d_in[7];
    const float* fe_pw1_w = (const float*)d_in[8];
    const float* fe_pw1_b = (const float*)d_in[9];
    const float* fe_pw2_w = (const float*)d_in[10];
    const float* fe_pw2_b = (const float*)d_in[11];
    const float* fe_scale = (const float*)d_in[12];
    const float* fe_out_w = (const float*)d_in[13];
    const float* fe_out_b = (const float*)d_in[14];
    const float* mag_w    = (const float*)d_in[15];
    const float* mag_b    = (const float*)d_in[16];
    const float* rv_in_w  = (const float*)d_in[17];
    const float* rv_in_b  = (const float*)d_in[18];
    const float* dcc_w    = (const float*)d_in[19];
    const float* dcc_b    = (const float*)d_in[20];
    const float* rv_out_w = (const float*)d_in[21];
    const float* rv_out_b = (const float*)d_in[22];
    float* outp = (float*)d_out;

    float* ws = (float*)d_ws;
    float* xbuf  = ws;                      // [4][256][256]
    float* hbuf  = xbuf  + (size_t)NB * CF * LF;
    float* h2buf = hbuf  + (size_t)NB * CF * LF;   // [4][512][256]
    float* featb = h2buf + (size_t)NB * HF * LF;
    float* magb  = featb + (size_t)NB * CF * LF;   // [4][48][256]
    float* csum  = magb  + (size_t)NB * NHARM * LF;
    float* cpre  = csum  + NB * 256;
    float* waveb = cpre  + NB * 256;               // [4][LW]
    float* rA    = waveb + (size_t)NB * LW;        // [4][16][LW]
    float* rB    = rA    + (size_t)NB * 16 * LW;
    float* partb = rB    + (size_t)NB * 16 * LW;
    float* meanb = partb + NB * 240;

    // ---- FeatureExtractor ----
    gemm_wmma_k<0, false, 80><<<dim3(CF / 16, 16, NB), 32, 0, stream>>>(
        fe_in_w, fe_in_b, mel, xbuf, nullptr, nullptr, CF);

    float* curX = xbuf;
    float* nxtX = hbuf;
    for (int i = 0; i < NLAY; ++i) {
        dwnorm_k<<<dim3(LF, NB), 256, 0, stream>>>(
            curX, nxtX, fe_dw_w + (size_t)i * CF * 7, fe_dw_b + i * CF,
            fe_ns + i * CF, fe_nb + i * CF);
        gemm_wmma_k<1, false, CF><<<dim3(HF / 16, 16, NB), 32, 0, stream>>>(
            fe_pw1_w + (size_t)i * HF * CF, fe_pw1_b + i * HF, nxtX, h2buf,
            nullptr, nullptr, HF);
        gemm_wmma_k<0, true, HF><<<dim3(CF / 16, 16, NB), 32, 0, stream>>>(
            fe_pw2_w + (size_t)i * CF * HF, fe_pw2_b + i * CF, h2buf, nxtX,
            curX, fe_scale + i * CF, CF);
        float* t = curX; curX = nxtX; nxtX = t;
    }
    gemm_wmma_k<0, false, CF><<<dim3(CF / 16, 16, NB), 32, 0, stream>>>(
        fe_out_w, fe_out_b, curX, featb, nullptr, nullptr, CF);

    // ---- magnitudes ----
    gemm_wmma_k<2, false, CF><<<dim3(NHARM / 16, 16, NB), 32, 0, stream>>>(
        mag_w, mag_b, featb, magb, nullptr, nullptr, NHARM);

    // ---- phase cumsum + synthesis ----
    f0_chunksum_k<<<dim3(256, NB), 256, 0, stream>>>(f0, csum);
    f0_prefix_k<<<NB, 256, 0, stream>>>(csum, cpre);
    synth_k<<<dim3(256, NB), 256, 0, stream>>>(f0, magb, cpre, waveb);

    // ---- reverb ----
    {
        const size_t tot = (size_t)NB * 16 * LW;
        rv_in_k<<<(unsigned)(tot / 256), 256, 0, stream>>>(waveb, rv_in_w, rv_in_b, rA);
    }
    reverb_wmma_k<1><<<dim3(LW / RTS, NB), 256, 0, stream>>>(
        rA, rB, dcc_w + 0 * 2560, dcc_b + 0 * 16);
    reverb_wmma_k<2><<<dim3(LW / RTS, NB), 256, 0, stream>>>(
        rB, rA, dcc_w + 1 * 2560, dcc_b + 1 * 16);
    reverb_wmma_k<4><<<dim3(LW / RTS, NB), 256, 0, stream>>>(
        rA, rB, dcc_w + 2 * 2560, dcc_b + 2 * 16);
    reverb_wmma_k<8><<<dim3(LW / RTS, NB), 256, 0, stream>>>(
        rB, rA, dcc_w + 3 * 2560, dcc_b + 3 * 16);
    rv_out_k<<<(unsigned)(((size_t)NB * LW) / 256), 256, 0, stream>>>(
        rA, rv_out_w, rv_out_b, outp);

    // ---- mean subtraction ----
    mean_part_k<<<dim3(240, NB), 256, 0, stream>>>(outp, partb);
    mean_final_k<<<NB, 256, 0, stream>>>(partb, meanb);
    sub_mean_k<<<(unsigned)(((size_t)NB * LW) / 256), 256, 0, stream>>>(outp, meanb);
}